// OrkyMultiHeadHydra_19258633356047
// MI455X (gfx1250) — compile-verified
//
#include <hip/hip_runtime.h>
#include <math.h>

// ---------------------------------------------------------------------------
// OrkyMultiHeadHydra on MI455X (gfx1250, wave32, WMMA).
// All matmuls use V_WMMA_F32_16X16X4_F32 (fp32 A/B, fp32 accum) to match the
// fp32 reference (top-k threshold is precision sensitive) while running on
// the matrix pipe (2048 FLOP/instr). Attention keeps the full 16x4096 score
// row in LDS (263KB of the 320KB/WGP) so the sparse head can do an exact
// radix-select for the k-th (409th) largest score before the softmax.
// ---------------------------------------------------------------------------

typedef __attribute__((ext_vector_type(2))) float v2f;
typedef __attribute__((ext_vector_type(8))) float v8f;

#if defined(__HIP_DEVICE_COMPILE__)
#if !__has_builtin(__builtin_amdgcn_wmma_f32_16x16x4_f32)
#error "missing __builtin_amdgcn_wmma_f32_16x16x4_f32 on device toolchain"
#endif
#endif

__device__ __forceinline__ v8f wmma4(v2f a, v2f b, v8f c) {
#if defined(__HIP_DEVICE_COMPILE__)
  // 8-arg form: (neg_a, A, neg_b, B, c_mod, C, reuse_a, reuse_b)
  return __builtin_amdgcn_wmma_f32_16x16x4_f32(false, a, false, b, (short)0, c,
                                               false, false);
#else
  (void)a; (void)b;
  return c;  // host pass only needs to parse this
#endif
}

__device__ __forceinline__ float gelu_exact(float x) {
  return 0.5f * x * (1.0f + erff(x * 0.7071067811865476f));
}

// order-preserving float -> uint key (for radix select of k-th largest)
__device__ __forceinline__ unsigned ordkey(float f) {
  unsigned u = __float_as_uint(f);
  return (u & 0x80000000u) ? ~u : (u | 0x80000000u);
}

constexpr int SROW = 4100;   // LDS row stride (dwords): conflict-free frags
constexpr int KSEL = 409;    // int(4096 * 0.1)

enum { F_BIAS = 1, F_GELU = 2, F_ACCUM = 4 };

// ---------------------------------------------------------------------------
// Generic fp32 WMMA GEMM:  C[M x N] (+)= A[M x K] @ W[K x N] (+ bias)(gelu)
// grid = (M/16, N/64), block = 128 (4 waves, one 16x16 N-subtile per wave).
// A-fragments: per-lane contiguous float2 (row ml, cols k+2*hi .. +1).
// B-fragments: two strided scalars W[k+2hi][n], W[k+2hi+1][n].
// All operands stay hot in L2 (192MB) / WGP$; compute-bound workload.
// ---------------------------------------------------------------------------
template <int FLAGS>
__global__ __launch_bounds__(128) void gemm_wmma(
    const float* __restrict__ A, int lda, const float* __restrict__ W, int ldw,
    const float* __restrict__ bias, float* __restrict__ C, int ldc, int K) {
  const int lane = threadIdx.x & 31;
  const int wave = threadIdx.x >> 5;
  const int ml = lane & 15;
  const int hi = lane >> 4;
  const int m0 = blockIdx.x * 16;
  const int n0 = blockIdx.y * 64 + wave * 16;

  const float* arow = A + (size_t)(m0 + ml) * lda + 2 * hi;
  const float* wcol = W + (size_t)(2 * hi) * ldw + n0 + ml;

  v8f acc = {0.f, 0.f, 0.f, 0.f, 0.f, 0.f, 0.f, 0.f};
  if (FLAGS & F_ACCUM) {
#pragma unroll
    for (int v = 0; v < 8; ++v)
      acc[v] = C[(size_t)(m0 + v + 8 * hi) * ldc + n0 + ml];
  }

#pragma unroll 8
  for (int k = 0; k < K; k += 4) {
    v2f a = *(const v2f*)(arow + k);
    v2f b;
    b.x = wcol[(size_t)k * ldw];
    b.y = wcol[(size_t)(k + 1) * ldw];
    acc = wmma4(a, b, acc);
  }

  const float bv = (FLAGS & F_BIAS) ? bias[n0 + ml] : 0.f;
#pragma unroll
  for (int v = 0; v < 8; ++v) {
    float x = acc[v] + bv;
    if (FLAGS & F_GELU) x = gelu_exact(x);
    C[(size_t)(m0 + v + 8 * hi) * ldc + n0 + ml] = x;
  }
}

// ---------------------------------------------------------------------------
// Attention (full or sparse-top-k) over one 16-query tile per workgroup.
// qkv: [8192 x 768] rows (q|k|v each 256 cols). Output written into the
// combined activation buffer at a column offset (ldo = 1024).
// Block = 256 threads (8 waves). LDS: scores 16 x SROW fp32 (~263KB).
// ---------------------------------------------------------------------------
template <bool SPARSE>
__global__ __launch_bounds__(256) void attn_wmma(const float* __restrict__ qkv,
                                                 float* __restrict__ outp,
                                                 int ldo) {
  extern __shared__ float smem[];
  float* sc = smem;                    // 16 * SROW scores/probs
  float* redf = smem + 16 * SROW;      // 256 reduction scratch
  float* rmaxs = redf + 256;           // 16
  float* rinvs = rmaxs + 16;           // 16
  __shared__ unsigned pfx[16];         // radix-select prefixes

  const int lane = threadIdx.x & 31;
  const int wave = threadIdx.x >> 5;
  const int ml = lane & 15;
  const int hi = lane >> 4;
  const int bidx = blockIdx.y;
  const int m0g = bidx * 4096 + blockIdx.x * 16;  // global row of query tile

  const float* Qr = qkv + (size_t)(m0g + ml) * 768 + 2 * hi;
  const float* Kb = qkv + (size_t)bidx * 4096 * 768 + 256;
  const float* Vb = qkv + (size_t)bidx * 4096 * 768 + 512;

  // Preload Q tile as 64 A-fragments (128 VGPRs) -- reused for all 256 ktiles
  v2f a[64];
#pragma unroll
  for (int j = 0; j < 64; ++j) a[j] = *(const v2f*)(Qr + 4 * j);

  // ---- scores = scale * Q @ K^T, materialized in LDS ----
  for (int nt = wave; nt < 256; nt += 8) {
    const float* krow = Kb + (size_t)(nt * 16 + ml) * 768 + 2 * hi;
    v8f acc = {0.f, 0.f, 0.f, 0.f, 0.f, 0.f, 0.f, 0.f};
#pragma unroll
    for (int j = 0; j < 64; ++j) {
      v2f b = *(const v2f*)(krow + 4 * j);  // K^T frag: contiguous float2
      acc = wmma4(a[j], b, acc);
    }
    const int nc = nt * 16 + ml;
#pragma unroll
    for (int v = 0; v < 8; ++v)
      sc[(v + 8 * hi) * SROW + nc] = acc[v] * 0.0625f;  // 1/sqrt(256)
  }
  __syncthreads();

  // ---- row-wise softmax (+ exact top-k threshold for sparse head) ----
  const int row = threadIdx.x >> 4;  // 16 threads per row
  const int idx = threadIdx.x & 15;
  float* srow = sc + row * SROW;

  float mx = -3.402823466e38f;
  for (int c = idx; c < 4096; c += 16) mx = fmaxf(mx, srow[c]);
  redf[threadIdx.x] = mx;
  __syncthreads();
  if (idx == 0) {
    float m2 = redf[row * 16];
    for (int i = 1; i < 16; ++i) m2 = fmaxf(m2, redf[row * 16 + i]);
    rmaxs[row] = m2;
  }
  __syncthreads();

  unsigned thr = 0u;
  if (SPARSE) {
    // radix select: largest X with count(key >= X) >= KSEL  ==  key of the
    // KSEL-th largest score; then keep scores whose key >= X (ties kept,
    // matching the reference's  scores >= kth  semantics).
    if (idx == 0) pfx[row] = 0u;
    __syncthreads();
    for (int bit = 31; bit >= 0; --bit) {
      const unsigned cand = pfx[row] | (1u << bit);
      int cnt = 0;
      for (int c = idx; c < 4096; c += 16)
        cnt += (ordkey(srow[c]) >= cand) ? 1 : 0;
      redf[threadIdx.x] = (float)cnt;
      __syncthreads();
      if (idx == 0) {
        int t = 0;
        for (int i = 0; i < 16; ++i) t += (int)redf[row * 16 + i];
        if (t >= KSEL) pfx[row] = cand;
      }
      __syncthreads();
    }
    thr = pfx[row];
  }

  const float rm = rmaxs[row];
  float sum = 0.f;
  for (int c = idx; c < 4096; c += 16) {
    float s = srow[c];
    float e = SPARSE ? ((ordkey(s) >= thr) ? __expf(s - rm) : 0.f)
                     : __expf(s - rm);
    srow[c] = e;  // unnormalized probs; normalize in P@V epilogue
    sum += e;
  }
  redf[threadIdx.x] = sum;
  __syncthreads();
  if (idx == 0) {
    float t = 0.f;
    for (int i = 0; i < 16; ++i) t += redf[row * 16 + i];
    rinvs[row] = 1.0f / t;
  }
  __syncthreads();

  // ---- out = (P @ V) * rinv ----  (A-frags from LDS, conflict-free stride)
  const float* prow = sc + ml * SROW + 2 * hi;
  for (int nt = wave * 2; nt < wave * 2 + 2; ++nt) {
    v8f acc = {0.f, 0.f, 0.f, 0.f, 0.f, 0.f, 0.f, 0.f};
    const float* vcol = Vb + (size_t)(2 * hi) * 768 + nt * 16 + ml;
#pragma unroll 4
    for (int j = 0; j < 1024; ++j) {
      v2f af = *(const v2f*)(prow + 4 * j);
      v2f bf;
      bf.x = vcol[(size_t)(4 * j) * 768];
      bf.y = vcol[(size_t)(4 * j + 1) * 768];
      acc = wmma4(af, bf, acc);
    }
#pragma unroll
    for (int v = 0; v < 8; ++v) {
      const int r = v + 8 * hi;
      outp[(size_t)(m0g + r) * ldo + nt * 16 + ml] = acc[v] * rinvs[r];
    }
  }
}

// ---------------------------------------------------------------------------
// Memory head retrieval: one wave per 16-row tile.
// scores = x_proj @ bank^T (16x64), softmax, retrieved = P @ bank (16x256).
// ---------------------------------------------------------------------------
__global__ __launch_bounds__(32) void memhead_wmma(
    const float* __restrict__ xproj, const float* __restrict__ bank,
    float* __restrict__ retr) {
  __shared__ float pr[16 * 68];
  const int lane = threadIdx.x;
  const int ml = lane & 15;
  const int hi = lane >> 4;
  const int m0 = blockIdx.x * 16;

  const float* ar = xproj + (size_t)(m0 + ml) * 256 + 2 * hi;
  v2f a[64];
#pragma unroll
  for (int j = 0; j < 64; ++j) a[j] = *(const v2f*)(ar + 4 * j);

  for (int nt = 0; nt < 4; ++nt) {
    const float* br = bank + (size_t)(nt * 16 + ml) * 256 + 2 * hi;
    v8f acc = {0.f, 0.f, 0.f, 0.f, 0.f, 0.f, 0.f, 0.f};
#pragma unroll
    for (int j = 0; j < 64; ++j) {
      v2f b = *(const v2f*)(br + 4 * j);  // bank^T frag: contiguous float2
      acc = wmma4(a[j], b, acc);
    }
#pragma unroll
    for (int v = 0; v < 8; ++v) pr[(v + 8 * hi) * 68 + nt * 16 + ml] = acc[v];
  }
  __syncthreads();

  if (lane < 16) {
    float* rr = pr + lane * 68;
    float mx = rr[0];
    for (int i = 1; i < 64; ++i) mx = fmaxf(mx, rr[i]);
    float s = 0.f;
    for (int i = 0; i < 64; ++i) {
      float e = __expf(rr[i] - mx);
      rr[i] = e;
      s += e;
    }
    const float inv = 1.0f / s;
    for (int i = 0; i < 64; ++i) rr[i] *= inv;
  }
  __syncthreads();

  const float* prow = pr + ml * 68 + 2 * hi;
  for (int nt = 0; nt < 16; ++nt) {
    v8f acc = {0.f, 0.f, 0.f, 0.f, 0.f, 0.f, 0.f, 0.f};
    const float* bcol = bank + (size_t)(2 * hi) * 256 + nt * 16 + ml;
#pragma unroll
    for (int j = 0; j < 16; ++j) {
      v2f af = *(const v2f*)(prow + 4 * j);
      v2f bf;
      bf.x = bcol[(size_t)(4 * j) * 256];
      bf.y = bcol[(size_t)(4 * j + 1) * 256];
      acc = wmma4(af, bf, acc);
    }
#pragma unroll
    for (int v = 0; v < 8; ++v)
      retr[(size_t)(m0 + v + 8 * hi) * 256 + nt * 16 + ml] = acc[v];
  }
}

// ---------------------------------------------------------------------------
// Orchestration. Scratch layout (floats), total 18,874,368 f = 75.5 MB:
//   [0        .. 6291456)  qkv       (8192x768), later reused as h2 (8192x512)
//   [6291456  .. 8388608)  x_proj    (8192x256), later reused as h1
//   [8388608  ..10485760)  retrieved (8192x256)
//   [10485760 ..18874368)  combined  (8192x1024), heads write column slices
// ---------------------------------------------------------------------------
extern "C" void kernel_launch(void* const* d_in, const int* in_sizes, int n_in,
                              void* d_out, int out_size, void* d_ws,
                              size_t ws_size, hipStream_t stream) {
  (void)in_sizes; (void)n_in; (void)out_size; (void)ws_size;
  const float* x            = (const float*)d_in[0];
  const float* local_qkv_w  = (const float*)d_in[1];
  const float* sparse_qkv_w = (const float*)d_in[2];
  const float* memory_bank  = (const float*)d_in[3];
  const float* memory_proj_w= (const float*)d_in[4];
  const float* mem_out_w    = (const float*)d_in[5];
  const float* mem_out_b    = (const float*)d_in[6];
  const float* pred_in_w    = (const float*)d_in[7];
  const float* pred_in_b    = (const float*)d_in[8];
  const float* pred1_w      = (const float*)d_in[9];
  const float* pred1_b      = (const float*)d_in[10];
  const float* pred2_w      = (const float*)d_in[11];
  const float* pred2_b      = (const float*)d_in[12];
  const float* warboss_w    = (const float*)d_in[13];
  const float* warboss_b    = (const float*)d_in[14];
  float* out = (float*)d_out;
  float* ws  = (float*)d_ws;

  float* qkv  = ws;
  float* xprj = ws + 6291456;
  float* retr = ws + 8388608;
  float* comb = ws + 10485760;
  float* h1 = xprj;  // safe: consumed before reuse (in-order stream)
  float* h2 = qkv;

  const size_t attn_lds = (size_t)(16 * SROW + 256 + 32) * sizeof(float);
  (void)hipFuncSetAttribute(
      reinterpret_cast<const void*>(&attn_wmma<false>),
      hipFuncAttributeMaxDynamicSharedMemorySize, (int)attn_lds);
  (void)hipFuncSetAttribute(
      reinterpret_cast<const void*>(&attn_wmma<true>),
      hipFuncAttributeMaxDynamicSharedMemorySize, (int)attn_lds);

  const dim3 blk(128);
  // local head
  gemm_wmma<0><<<dim3(512, 12), blk, 0, stream>>>(x, 1024, local_qkv_w, 768,
                                                  nullptr, qkv, 768, 1024);
  attn_wmma<false><<<dim3(256, 2), 256, attn_lds, stream>>>(qkv, comb + 0, 1024);
  // sparse head (reuses qkv buffer)
  gemm_wmma<0><<<dim3(512, 12), blk, 0, stream>>>(x, 1024, sparse_qkv_w, 768,
                                                  nullptr, qkv, 768, 1024);
  attn_wmma<true><<<dim3(256, 2), 256, attn_lds, stream>>>(qkv, comb + 256, 1024);
  // memory head
  gemm_wmma<0><<<dim3(512, 4), blk, 0, stream>>>(x, 1024, memory_proj_w, 256,
                                                 nullptr, xprj, 256, 1024);
  memhead_wmma<<<dim3(512), dim3(32), 0, stream>>>(xprj, memory_bank, retr);
  gemm_wmma<F_BIAS><<<dim3(512, 4), blk, 0, stream>>>(
      xprj, 256, mem_out_w, 256, mem_out_b, comb + 512, 1024, 256);
  gemm_wmma<F_ACCUM><<<dim3(512, 4), blk, 0, stream>>>(
      retr, 256, mem_out_w + 256 * 256, 256, nullptr, comb + 512, 1024, 256);
  // prediction head
  gemm_wmma<F_BIAS><<<dim3(512, 4), blk, 0, stream>>>(
      x, 1024, pred_in_w, 256, pred_in_b, h1, 256, 1024);
  gemm_wmma<F_BIAS | F_GELU><<<dim3(512, 8), blk, 0, stream>>>(
      h1, 256, pred1_w, 512, pred1_b, h2, 512, 256);
  gemm_wmma<F_BIAS><<<dim3(512, 4), blk, 0, stream>>>(
      h2, 512, pred2_w, 256, pred2_b, comb + 768, 1024, 512);
  // final projection: out = combined @ warboss_w + b
  gemm_wmma<F_BIAS><<<dim3(512, 16), blk, 0, stream>>>(
      comb, 1024, warboss_w, 1024, warboss_b, out, 1024, 1024);
}